// TTC_17325898072472
// MI455X (gfx1250) — compile-verified
//
#include <hip/hip_runtime.h>
#include <stdint.h>

#define BB   16
#define LL   16
#define HH   320
#define WW   960
#define NPIXW (HH * WW)
#define MIMI 100      // mi*mi samples per box
#define NS   112      // padded sample slots: 7 waves * 16
#define TMAX 70       // max box side (len < 70)

typedef float    v2f  __attribute__((ext_vector_type(2)));
typedef float    v8f  __attribute__((ext_vector_type(8)));
typedef uint32_t u32x4 __attribute__((ext_vector_type(4)));
typedef int      i32x4 __attribute__((ext_vector_type(4)));
typedef int      i32x8 __attribute__((ext_vector_type(8)));

__device__ __forceinline__ uint32_t mix32(uint32_t x) {
  x ^= x >> 16; x *= 0x7feb352du;
  x ^= x >> 15; x *= 0x846ca68bu;
  x ^= x >> 16;
  return x;
}

__device__ __forceinline__ uint32_t rfl(uint32_t v) {
  return (uint32_t)__builtin_amdgcn_readfirstlane((int)v);
}

__global__ __launch_bounds__(256) void TTC_box_sample_kernel(
    const float* __restrict__ bl,    // [B, L, 10]
    const float* __restrict__ flow,  // [B, 2, H, W]
    const float* __restrict__ dc,    // [B, 1, H, W]
    float* __restrict__ out)         // [B*L, 7, mi, mi]
{
  __shared__ float    tile[TMAX * TMAX];       // flow[b,0] box tile
  __shared__ uint32_t validList[TMAX * TMAX];  // compacted valid pixel ids (y*W+x)
  __shared__ int      vcnt;
  __shared__ uint32_t sampIdx[NS];
  __shared__ float    sxs[NS], sys[NS], sfx[NS], sfy[NS], sdc[NS];

  const int box  = blockIdx.x;          // b*L + l
  const int b    = box / LL;
  const int tid  = threadIdx.x;
  const unsigned lane = tid & 31u;
  const unsigned wave = tid >> 5;

  // ---- box parameters (uniform across block) ----
  const float maxx0 = bl[box * 10 + 2];
  const float minx0 = bl[box * 10 + 3];
  const float maxy0 = bl[box * 10 + 4];
  const float miny0 = bl[box * 10 + 5];
  const float maxx1 = bl[box * 10 + 6];
  const float minx1 = bl[box * 10 + 7];
  const float maxy1 = bl[box * 10 + 8];
  const float miny1 = bl[box * 10 + 9];

  // integer pixel range strictly inside (minx0, maxx0) x (miny0, maxy0)
  int xlo = (int)floorf(minx0) + 1;
  int ylo = (int)floorf(miny0) + 1;
  int xhi = (int)ceilf(maxx0) - 1;
  int yhi = (int)ceilf(maxy0) - 1;
  xlo = max(xlo, 0); ylo = max(ylo, 0);
  xhi = min(xhi, WW - 1); yhi = min(yhi, HH - 1);
  int wBox = min(max(xhi - xlo + 1, 1), TMAX);
  int hBox = min(max(yhi - ylo + 1, 1), TMAX);

  if (tid == 0) vcnt = 0;

  const uint64_t gaddr =
      (uint64_t)(uintptr_t)(flow + ((size_t)b * 2) * NPIXW + (size_t)ylo * WW + xlo);

  // ---- stage the 2D box tile of flow[b,0] into LDS via the Tensor Data Mover ----
#if __has_builtin(__builtin_amdgcn_tensor_load_to_lds)
  if (wave == 0) {
    const uint32_t laddr = (uint32_t)(uintptr_t)(&tile[0]);
    u32x4 g0;
    g0[0] = rfl(1u);                                   // count=1, user descriptor
    g0[1] = rfl(laddr);                                // lds_addr (bytes)
    g0[2] = rfl((uint32_t)gaddr);                      // global_addr[31:0] (tile start)
    g0[3] = rfl(((uint32_t)(gaddr >> 32) & 0x01FFFFFFu) | (2u << 30)); // addr[56:32] | type=2

    i32x8 g1;
    g1[0] = (int)rfl(2u << 16);                                            // data_size=4B
    g1[1] = (int)rfl(((uint32_t)WW & 0xFFFFu) << 16);                      // tensor_dim0[15:0]
    g1[2] = (int)rfl(((uint32_t)WW >> 16) | (((uint32_t)HH & 0xFFFFu) << 16)); // dim0 hi | dim1 lo
    g1[3] = (int)rfl(((uint32_t)HH >> 16) | ((uint32_t)wBox << 16));       // dim1 hi | tile_dim0
    g1[4] = (int)rfl((uint32_t)hBox);                                      // tile_dim1 | tile_dim2=0
    g1[5] = (int)rfl((uint32_t)WW);                                        // tensor_dim0_stride lo32
    g1[6] = 0;                                                             // stride0 hi | stride1 lo
    g1[7] = 0;

    i32x4 gz = {0, 0, 0, 0};
#if defined(__clang_major__) && __clang_major__ >= 23
    i32x8 gz8 = {0, 0, 0, 0, 0, 0, 0, 0};
    __builtin_amdgcn_tensor_load_to_lds(g0, g1, gz, gz, gz8, 0);
#else
    __builtin_amdgcn_tensor_load_to_lds(g0, g1, gz, gz, 0);
#endif
  }
#if __has_builtin(__builtin_amdgcn_s_wait_tensorcnt)
  __builtin_amdgcn_s_wait_tensorcnt(0);
#else
  asm volatile("s_wait_tensorcnt 0" ::: "memory");
#endif
#else
  // Fallback: cooperative strided tile load
  for (int r = tid; r < wBox * hBox; r += 256) {
    tile[r] = flow[((size_t)b * 2) * NPIXW + (size_t)(ylo + r / wBox) * WW + (xlo + r % wBox)];
  }
#endif
  __syncthreads();

  // ---- compact valid pixels (inside box interior, flow != 0) ----
  const int nPix = wBox * hBox;
  for (int r = tid; r < nPix; r += 256) {
    const int xx = xlo + (r % wBox);
    const int yy = ylo + (r / wBox);
    const float f0 = tile[r];
    const bool ok = ((float)xx > minx0) && ((float)xx < maxx0) &&
                    ((float)yy > miny0) && ((float)yy < maxy0) && (f0 != 0.0f);
    if (ok) {
      const int p = atomicAdd(&vcnt, 1);
      validList[p] = (uint32_t)(yy * WW + xx);
    }
  }
  __syncthreads();
  const int cnt = vcnt;

  // ---- categorical sampling: uniform-over-valid == argmax of raw hash (monotone Gumbel) ----
  const uint32_t seed = 0x9E3779B9u * (uint32_t)(box + 1);
  for (int s = (int)wave; s < MIMI; s += 8) {
    uint32_t bestH = 0u;
    uint32_t bestI = 0xFFFFFFFFu;
    for (uint32_t i = lane; i < (uint32_t)cnt; i += 32u) {
      const uint32_t h = mix32(seed ^ ((uint32_t)s * 0x85EBCA6Bu) ^ (i * 0xC2B2AE35u));
      if (bestI == 0xFFFFFFFFu || h > bestH) { bestH = h; bestI = i; }
    }
    // wave32 max-reduction
    for (int off = 16; off > 0; off >>= 1) {
      const uint32_t oh = (uint32_t)__shfl_xor((int)bestH, off, 32);
      const uint32_t oi = (uint32_t)__shfl_xor((int)bestI, off, 32);
      if (oi != 0xFFFFFFFFu && (bestI == 0xFFFFFFFFu || oh > bestH)) { bestH = oh; bestI = oi; }
    }
    if (lane == 0) {
      uint32_t gidx;
      if (cnt > 0 && bestI != 0xFFFFFFFFu) gidx = validList[bestI];
      else gidx = mix32(seed ^ (uint32_t)s) % (uint32_t)NPIXW;  // degenerate box
      sampIdx[s] = gidx;
    }
  }
  __syncthreads();

  // ---- gather flow/dc at sampled pixels, stage per-sample values ----
  if (tid < NS) {
    float xs = 0.f, ys = 0.f, fx = 0.f, fy = 0.f, dv = 0.f;
    if (tid < MIMI) {
      const uint32_t gi = sampIdx[tid];
      xs = (float)(gi % (uint32_t)WW);
      ys = (float)(gi / (uint32_t)WW);
      const size_t base0 = ((size_t)b * 2) * NPIXW;
      fx = flow[base0 + gi];
      fy = flow[base0 + NPIXW + gi];
      dv = dc[(size_t)b * NPIXW + gi];
    }
    sxs[tid] = xs; sys[tid] = ys; sfx[tid] = fx; sfy[tid] = fy; sdc[tid] = dv;
  }
  __syncthreads();

  const size_t obase = (size_t)box * 7 * MIMI;

  // ---- channels 4..6: elementwise ----
  if (tid < MIMI) {
    out[obase + 4 * MIMI + tid] = sfx[tid] * 0.125f;
    out[obase + 5 * MIMI + tid] = sfy[tid] * 0.125f;
    out[obase + 6 * MIMI + tid] = sdc[tid];
  }

  // ---- channels 0..3: affine map as f32 WMMA  D(16x16) = A(16x4) x B(4x16) ----
  const float lx0 = maxx0 - minx0, ly0 = maxy0 - miny0;
  const float lx1 = maxx1 - minx1, ly1 = maxy1 - miny1;
  const float sx8 = lx1 / lx0 * 0.125f;
  const float sy8 = ly1 / ly0 * 0.125f;
  const float tx8 = (minx1 - minx0 * lx1 / lx0) * 0.125f;
  const float ty8 = (miny1 - miny0 * ly1 / ly0) * 0.125f;

  if (wave < 7) {   // 7 waves x 16 sample columns = 112 slots; EXEC all-ones per wave
    const int  s0 = (int)wave * 16;
    const int  m  = (int)(lane & 15u);
    const bool hi = lane >= 16u;

    // A 16x4 f32: VGPR0 = K0(lo)/K2(hi), VGPR1 = K1(lo)/K3(hi); rows m=0..3 carry coefs
    v2f a; a.x = 0.f; a.y = 0.f;
    if (!hi) {
      if (m == 0) a.x = 0.125f; else if (m == 2) a.x = sx8;
      if (m == 1) a.y = 0.125f; else if (m == 3) a.y = sy8;
    } else {
      if (m == 2) a.x = tx8;   else if (m == 3) a.x = ty8;
    }
    // B 4x16 f32: column n = s0+m holds [x, y, 1, 0]
    v2f bm;
    if (!hi) { bm.x = sxs[s0 + m]; bm.y = sys[s0 + m]; }
    else     { bm.x = 1.0f;        bm.y = 0.0f; }

    v8f c = {};
    c = __builtin_amdgcn_wmma_f32_16x16x4_f32(
        /*neg_a=*/false, a, /*neg_b=*/false, bm,
        /*c_mod=*/(short)0, c, /*reuse_a=*/false, /*reuse_b=*/false);

    if (!hi) {
      const int s = s0 + m;
      if (s < MIMI) {
        out[obase + 0 * MIMI + s] = c[0];  // M=0 -> VGPR0, lanes 0-15
        out[obase + 1 * MIMI + s] = c[1];
        out[obase + 2 * MIMI + s] = c[2];
        out[obase + 3 * MIMI + s] = c[3];
      }
    }
  }
}

extern "C" void kernel_launch(void* const* d_in, const int* in_sizes, int n_in,
                              void* d_out, int out_size, void* d_ws, size_t ws_size,
                              hipStream_t stream) {
  (void)in_sizes; (void)n_in; (void)d_ws; (void)ws_size; (void)out_size;
  const float* bl   = (const float*)d_in[0];
  const float* flow = (const float*)d_in[1];
  const float* dc   = (const float*)d_in[2];
  // d_in[3] (boxnum) is unused by the reference math; d_in[4] (mi=10) hardcoded.
  float* out = (float*)d_out;
  TTC_box_sample_kernel<<<dim3(BB * LL), dim3(256), 0, stream>>>(bl, flow, dc, out);
}